// CausalGraphAttention_62457414419225
// MI455X (gfx1250) — compile-verified
//
#include <hip/hip_runtime.h>

#define N_NODES 8192
#define F 256
#define ALPHA 0.2f

typedef __attribute__((ext_vector_type(2))) float v2f;
typedef __attribute__((ext_vector_type(8))) float v8f;

__device__ __forceinline__ v8f wmma_f32_16x16x4(v2f a, v2f b, v8f c) {
  // D(16x16,f32) = A(16x4,f32) * B(4x16,f32) + C
  return __builtin_amdgcn_wmma_f32_16x16x4_f32(false, a, false, b, (short)0, c,
                                               false, false);
}

__device__ __forceinline__ float leaky(float z) { return z > 0.f ? z : ALPHA * z; }

#if __has_builtin(__builtin_amdgcn_s_wait_asynccnt)
#define WAIT_ASYNC(n) __builtin_amdgcn_s_wait_asynccnt(n)
#else
#define WAIT_ASYNC(n) asm volatile("s_wait_asynccnt %0" ::"i"(n) : "memory")
#endif

// Issue 8 x GLOBAL_LOAD_ASYNC_TO_LDS_B128: 128 bytes per lane, memory -> LDS
// directly (no VGPR round trip), tracked by ASYNCcnt.  The IOFFSET immediate
// advances BOTH the global and the LDS address (CDNA5 ISA).
__device__ __forceinline__ void issue_htile_async(const float* __restrict__ h,
                                                  int jb, float* lds_base,
                                                  int ldr, int ldc) {
  const float* gsrc = h + (long)(jb + ldr) * F + ldc;
  // flat shared-aperture address: low 32 bits == LDS byte offset
  const unsigned lds_off = (unsigned)(size_t)(lds_base + ldr * 264 + ldc);
  asm volatile(
      "global_load_async_to_lds_b128 %0, %1, off\n\t"
      "global_load_async_to_lds_b128 %0, %1, off offset:16\n\t"
      "global_load_async_to_lds_b128 %0, %1, off offset:32\n\t"
      "global_load_async_to_lds_b128 %0, %1, off offset:48\n\t"
      "global_load_async_to_lds_b128 %0, %1, off offset:64\n\t"
      "global_load_async_to_lds_b128 %0, %1, off offset:80\n\t"
      "global_load_async_to_lds_b128 %0, %1, off offset:96\n\t"
      "global_load_async_to_lds_b128 %0, %1, off offset:112"
      :: "v"(lds_off), "v"(gsrc)
      : "memory");
}

// ---------------------------------------------------------------------------
// Kernel 1: h = x @ W  (8192x256 = [8192x256]x[256x256]) via f32 WMMA.
// One wave per 16x16 output tile, K-loop in steps of 4.
// ---------------------------------------------------------------------------
__global__ __launch_bounds__(32) void k_gemm_h(const float* __restrict__ x,
                                               const float* __restrict__ W,
                                               float* __restrict__ h) {
  const int lane = threadIdx.x & 31;
  const int cidx = lane & 15;
  const int half = lane >> 4;
  const int off  = half * 2;
  const int rowb = blockIdx.x * 16;
  const int colb = blockIdx.y * 16;
  const int row  = rowb + cidx;
  const int col  = colb + cidx;

  v8f acc = {0.f, 0.f, 0.f, 0.f, 0.f, 0.f, 0.f, 0.f};
  for (int kb = 0; kb < F; kb += 4) {
    v2f a = *(const v2f*)&x[(long)row * F + kb + off];
    v2f b;
    b.x = W[(kb + off) * F + col];
    b.y = W[(kb + off + 1) * F + col];
    acc = wmma_f32_16x16x4(a, b, acc);
  }
#pragma unroll
  for (int v = 0; v < 8; ++v)
    h[(long)(rowb + v + 8 * half) * F + col] = acc[v];
}

// ---------------------------------------------------------------------------
// Kernel 2: per-row dot products: s_src = h.a1, s_dst = h.a2, cs = x.c
// ---------------------------------------------------------------------------
__global__ __launch_bounds__(256) void k_rowdots(
    const float* __restrict__ x, const float* __restrict__ h,
    const float* __restrict__ a1, const float* __restrict__ a2,
    const float* __restrict__ cvec, float* __restrict__ s_src,
    float* __restrict__ s_dst, float* __restrict__ cs) {
  const int lane = threadIdx.x & 31;
  const int row  = blockIdx.x * 8 + (threadIdx.x >> 5);
  const float* hrow = h + (long)row * F;
  const float* xrow = x + (long)row * F;
  float s1 = 0.f, s2 = 0.f, s3 = 0.f;
#pragma unroll
  for (int cb = 0; cb < F; cb += 32) {
    const int cc = cb + lane;
    const float hv = hrow[cc];
    const float xv = xrow[cc];
    s1 += hv * a1[cc];
    s2 += hv * a2[cc];
    s3 += xv * cvec[cc];
  }
#pragma unroll
  for (int o = 16; o > 0; o >>= 1) {
    s1 += __shfl_xor(s1, o);
    s2 += __shfl_xor(s2, o);
    s3 += __shfl_xor(s3, o);
  }
  if (lane == 0) {
    s_src[row] = s1;
    s_dst[row] = s2;
    cs[row]    = s3;
  }
}

// ---------------------------------------------------------------------------
// Kernel 3: maxs[0] = max(s_dst), maxs[1] = max(cs)  (for the row-max bound)
// ---------------------------------------------------------------------------
__global__ __launch_bounds__(256) void k_maxred(const float* __restrict__ s_dst,
                                                const float* __restrict__ cs,
                                                float* __restrict__ maxs) {
  __shared__ float smd[8];
  __shared__ float smc[8];
  const int tid = threadIdx.x, lane = tid & 31, wid = tid >> 5;
  float md = -3.4e38f, mc = -3.4e38f;
  for (int i = tid; i < N_NODES; i += 256) {
    md = fmaxf(md, s_dst[i]);
    mc = fmaxf(mc, cs[i]);
  }
#pragma unroll
  for (int o = 16; o > 0; o >>= 1) {
    md = fmaxf(md, __shfl_xor(md, o));
    mc = fmaxf(mc, __shfl_xor(mc, o));
  }
  if (lane == 0) { smd[wid] = md; smc[wid] = mc; }
  __syncthreads();
  if (tid == 0) {
    float a = smd[0], b = smc[0];
#pragma unroll
    for (int i = 1; i < 8; ++i) { a = fmaxf(a, smd[i]); b = fmaxf(b, smc[i]); }
    maxs[0] = a;
    maxs[1] = b;
  }
}

// ---------------------------------------------------------------------------
// Kernel 4: fused masked-softmax attention * h, one pass, f32 WMMA PV-GEMM,
// double-buffered LDS h-tiles filled by GLOBAL_LOAD_ASYNC_TO_LDS_B128 with
// prefetch distance 1 (next tile's async copies overlap this tile's WMMAs).
// Row max uses the analytic bound m_i = leaky(s_src+maxd)+max(0,maxc-cs_i):
// softmax is shift-invariant, so no rescaling / second pass is needed.
// ---------------------------------------------------------------------------
__global__ __launch_bounds__(128) void k_attn(
    const int* __restrict__ adj, const float* __restrict__ cw,
    const float* __restrict__ h, const float* __restrict__ s_src,
    const float* __restrict__ s_dst, const float* __restrict__ cs,
    const float* __restrict__ maxs, float* __restrict__ out) {
  __shared__ __align__(16) float lds_h[2][16 * 264];  // stride 264: bank-safe

  const int tid  = threadIdx.x;
  const int lane = tid & 31;
  const int wid  = tid >> 5;
  const int cidx = lane & 15;   // A-layout row index / B,C column index
  const int half = lane >> 4;
  const int off  = half * 2;    // K offset of this lane-half inside a K=4 step
  const int rowbase = blockIdx.x * 64 + wid * 16;
  const int rowA    = rowbase + cidx;  // row this lane owns in the A operand

  const float maxd = maxs[0];
  const float maxc = maxs[1];
  const float ss   = s_src[rowA];
  const float csr  = cs[rowA];
  const float m    = leaky(ss + maxd) + fmaxf(0.f, maxc - csr);  // >= row max

  const long   rb    = (long)rowA * N_NODES;
  const float* cwrow = cw + rb;
  const int*   adrow = adj + rb;

  v8f acc[16];
  const v8f vzero = {0.f, 0.f, 0.f, 0.f, 0.f, 0.f, 0.f, 0.f};
#pragma unroll
  for (int t = 0; t < 16; ++t) acc[t] = vzero;
  float lsum = 0.f;

  // cooperative staging indices: 128 threads x 128 B = one 16x256 f32 tile
  const int ldr = tid >> 3;        // tile row 0..15
  const int ldc = (tid & 7) * 32;  // column base (32 floats = 8 x b128)

  // prologue: start tile 0 into buffer 0
  issue_htile_async(h, 0, &lds_h[0][0], ldr, ldc);

  for (int idx = 0; idx < N_NODES / 16; ++idx) {
    const int jb  = idx * 16;
    const int cur = idx & 1;

    if (jb + 16 < N_NODES) {
      issue_htile_async(h, jb + 16, &lds_h[cur ^ 1][0], ldr, ldc);
      WAIT_ASYNC(8);   // 8 newer ops may remain in flight -> current tile done
    } else {
      WAIT_ASYNC(0);
    }
    __syncthreads();   // all waves' async copies for this tile are visible

    __builtin_prefetch(cwrow + jb + 16, 0, 0);  // global_prefetch_b8
    __builtin_prefetch(adrow + jb + 16, 0, 0);

    // ---- build probability tile P in WMMA A-operand layout --------------
    v2f a[4];
#pragma unroll
    for (int q = 0; q < 4; ++q) {
      const int j0 = jb + 4 * q + off;  // even -> 8B aligned
      const v2f sd  = *(const v2f*)&s_dst[j0];
      const v2f cj  = *(const v2f*)&cs[j0];
      const v2f cwv = *(const v2f*)&cwrow[j0];
      const int2 ad = *(const int2*)&adrow[j0];
      const float ex = leaky(ss + sd.x) + (cj.x - csr) * cwv.x;
      const float ey = leaky(ss + sd.y) + (cj.y - csr) * cwv.y;
      const float px = (ad.x > 0) ? __expf(ex - m) : 0.f;
      const float py = (ad.y > 0) ? __expf(ey - m) : 0.f;
      a[q].x = px;
      a[q].y = py;
      lsum += px + py;
    }

    // ---- acc += P(16x16) @ H(16x256): 16 N-tiles x 4 K-steps of WMMA ----
    const float* hbase = &lds_h[cur][0];
#pragma unroll
    for (int t = 0; t < 16; ++t) {
      const float* hb = hbase + t * 16 + cidx;
#pragma unroll
      for (int q = 0; q < 4; ++q) {
        v2f b;
        b.x = hb[(4 * q + off) * 264];
        b.y = hb[(4 * q + off + 1) * 264];
        acc[t] = wmma_f32_16x16x4(a[q], b, acc[t]);
      }
    }
    __syncthreads();   // tile consumed; next iteration may overwrite it
  }

  // lanes L and L+16 hold the two halves of row (L&15)'s sum
  lsum += __shfl_xor(lsum, 16);

  // redistribute row sums to match the C layout (row = v + 8*half)
  float linv[8];
#pragma unroll
  for (int v = 0; v < 8; ++v) {
    const float lv = __shfl(lsum, v + half * 8);
    linv[v] = 1.0f / fmaxf(lv, 1e-30f);
  }

#pragma unroll
  for (int t = 0; t < 16; ++t) {
#pragma unroll
    for (int v = 0; v < 8; ++v) {
      out[(long)(rowbase + v + 8 * half) * F + t * 16 + cidx] =
          acc[t][v] * linv[v];
    }
  }
}

// ---------------------------------------------------------------------------
extern "C" void kernel_launch(void* const* d_in, const int* in_sizes, int n_in,
                              void* d_out, int out_size, void* d_ws,
                              size_t ws_size, hipStream_t stream) {
  (void)in_sizes; (void)n_in; (void)out_size; (void)ws_size;
  const float* x   = (const float*)d_in[0];
  const int*   adj = (const int*)d_in[1];
  const float* cw  = (const float*)d_in[2];
  const float* W   = (const float*)d_in[3];
  const float* a1  = (const float*)d_in[4];
  const float* a2  = (const float*)d_in[5];
  const float* c   = (const float*)d_in[6];
  float* out = (float*)d_out;

  float* ws    = (float*)d_ws;            // workspace layout (floats):
  float* h     = ws;                      // [8192*256] projected features
  float* s_src = h + (size_t)N_NODES * F; // [8192]
  float* s_dst = s_src + N_NODES;         // [8192]
  float* cs    = s_dst + N_NODES;         // [8192]
  float* maxs  = cs + N_NODES;            // [2] max(s_dst), max(cs)

  k_gemm_h<<<dim3(N_NODES / 16, F / 16), 32, 0, stream>>>(x, W, h);
  k_rowdots<<<N_NODES / 8, 256, 0, stream>>>(x, h, a1, a2, c, s_src, s_dst, cs);
  k_maxred<<<1, 256, 0, stream>>>(s_dst, cs, maxs);
  k_attn<<<N_NODES / 64, 128, 0, stream>>>(adj, cw, h, s_src, s_dst, cs, maxs,
                                           out);
}